// QuantilePreprocessing_49589692400090
// MI455X (gfx1250) — compile-verified
//
#include <hip/hip_runtime.h>
#include <hip/hip_bf16.h>
#include <stdint.h>

#define QN    1000          // quantiles
#define QPAD  1024          // padded rows (pad with +inf -> branchless search)
#define FEAT  32            // features
#define LDS_BYTES (QPAD * FEAT * 4)   // 131072 B (fits: 320 KB LDS/WGP)

typedef int v4i_t __attribute__((ext_vector_type(4)));
typedef __attribute__((address_space(1))) v4i_t gas_v4i;   // global
typedef __attribute__((address_space(3))) v4i_t las_v4i;   // LDS

// ---------------------------------------------------------------------------
// erfinv(t) * sqrt(2), clamped to [-100,100]; Giles (2010) SP approximation.
// Poles at t = +/-1 produce inf in the reference -> clip gives +/-100; we
// special-case them so the polynomial's inf arithmetic can't flip the sign.
// ---------------------------------------------------------------------------
__device__ __forceinline__ float norm_inv_from_t(float t) {
    float w = -__logf((1.0f - t) * (1.0f + t));
    float p;
    if (w < 5.0f) {
        w -= 2.5f;
        p = 2.81022636e-08f;
        p = __fmaf_rn(p, w, 3.43273939e-07f);
        p = __fmaf_rn(p, w, -3.5233877e-06f);
        p = __fmaf_rn(p, w, -4.39150654e-06f);
        p = __fmaf_rn(p, w, 0.00021858087f);
        p = __fmaf_rn(p, w, -0.00125372503f);
        p = __fmaf_rn(p, w, -0.00417768164f);
        p = __fmaf_rn(p, w, 0.246640727f);
        p = __fmaf_rn(p, w, 1.50140941f);
    } else {
        w = __fsqrt_rn(w) - 3.0f;
        p = -0.000200214257f;
        p = __fmaf_rn(p, w, 0.000100950558f);
        p = __fmaf_rn(p, w, 0.00134934322f);
        p = __fmaf_rn(p, w, -0.00367342844f);
        p = __fmaf_rn(p, w, 0.00573950773f);
        p = __fmaf_rn(p, w, -0.0076224613f);
        p = __fmaf_rn(p, w, 0.00943887047f);
        p = __fmaf_rn(p, w, 1.00167406f);
        p = __fmaf_rn(p, w, 2.83297682f);
    }
    float r = p * t * 1.41421356237f;
    r = fminf(fmaxf(r, -100.0f), 100.0f);
    r = (t >=  1.0f) ?  100.0f : r;
    r = (t <= -1.0f) ? -100.0f : r;
    return r;
}

// epilogue shared by vector & scalar paths
__device__ __forceinline__ float finish(float xv, int cnt,
                                        const float* __restrict__ sq, int f) {
    int idx = cnt;
    idx = (idx < 1) ? 1 : idx;
    idx = (idx > QN - 1) ? QN - 1 : idx;
    idx -= 1;                                  // in [0, QN-2]
    float lo = sq[idx * FEAT + f];
    float hi = sq[idx * FEAT + FEAT + f];
    float diff = hi - lo;
    float interp = (diff == 0.0f) ? 0.5f : __fdividef(xv - lo, diff);
    float y = ((float)idx + interp) * (1.0f / (float)QN);
    y = fminf(fmaxf(y, 0.0f), 1.0f);
    float t = __fmaf_rn(2.0f, y, -1.0f);
    return norm_inv_from_t(t);
}

// ---------------------------------------------------------------------------
// 4 elements at once: per search step, issue 4 independent ds_loads
// back-to-back (one dscnt wait for all four), then 4 compare/updates.
// Row-major [QPAD][32] LDS layout -> bank = (f + 32*q) % 64 is distinct for
// the 32 distinct f's of a wave -> conflict-free for any per-lane q.
// ---------------------------------------------------------------------------
__device__ __forceinline__ void process_four(const float* __restrict__ x,
                                             float* __restrict__ out,
                                             const float* __restrict__ sq,
                                             int i, int T, int f) {
    // batched streaming loads (one loadcnt wait)
    float xv0 = __builtin_nontemporal_load(x + i);
    float xv1 = __builtin_nontemporal_load(x + i + T);
    float xv2 = __builtin_nontemporal_load(x + i + 2 * T);
    float xv3 = __builtin_nontemporal_load(x + i + 3 * T);

    int c0 = 0, c1 = 0, c2 = 0, c3 = 0;
#pragma unroll
    for (int step = 512; step >= 1; step >>= 1) {
        // 4 independent LDS loads issued together
        float v0 = sq[(c0 + step - 1) * FEAT + f];
        float v1 = sq[(c1 + step - 1) * FEAT + f];
        float v2 = sq[(c2 + step - 1) * FEAT + f];
        float v3 = sq[(c3 + step - 1) * FEAT + f];
        c0 += (v0 < xv0) ? step : 0;
        c1 += (v1 < xv1) ? step : 0;
        c2 += (v2 < xv2) ? step : 0;
        c3 += (v3 < xv3) ? step : 0;
    }

    float r0 = finish(xv0, c0, sq, f);
    float r1 = finish(xv1, c1, sq, f);
    float r2 = finish(xv2, c2, sq, f);
    float r3 = finish(xv3, c3, sq, f);

    __builtin_nontemporal_store(r0, out + i);
    __builtin_nontemporal_store(r1, out + i + T);
    __builtin_nontemporal_store(r2, out + i + 2 * T);
    __builtin_nontemporal_store(r3, out + i + 3 * T);
}

__device__ __forceinline__ void process_one(const float* __restrict__ x,
                                            float* __restrict__ out,
                                            const float* __restrict__ sq,
                                            int i, int f) {
    float xv = __builtin_nontemporal_load(x + i);
    int cnt = 0;
#pragma unroll
    for (int step = 512; step >= 1; step >>= 1) {
        float v = sq[(cnt + step - 1) * FEAT + f];
        if (v < xv) cnt += step;
    }
    __builtin_nontemporal_store(finish(xv, cnt, sq, f), out + i);
}

__global__ __launch_bounds__(256)
void QuantilePreprocessing_kernel(const float* __restrict__ x,
                                  const float* __restrict__ quant,
                                  float* __restrict__ out, int n) {
    extern __shared__ float sq[];   // [QPAD][FEAT] floats = 128 KB

    // ---- stage quantile table into LDS via gfx1250 async b128 copies ----
    {
        const float4* g4 = (const float4*)quant;
        float4*       l4 = (float4*)sq;
        for (int i = threadIdx.x; i < (QN * FEAT) / 4; i += blockDim.x) {
#if __has_builtin(__builtin_amdgcn_global_load_async_to_lds_b128)
            __builtin_amdgcn_global_load_async_to_lds_b128(
                (gas_v4i*)(uintptr_t)(g4 + i),
                (las_v4i*)(uintptr_t)(l4 + i),
                0, 0);
#else
            l4[i] = g4[i];
#endif
        }
        // pad rows [QN, QPAD) with +inf -> search needs no bounds checks
        for (int i = threadIdx.x; i < (QPAD - QN) * FEAT; i += blockDim.x) {
            sq[QN * FEAT + i] = __builtin_inff();
        }
    }
#if __has_builtin(__builtin_amdgcn_s_wait_asynccnt)
    __builtin_amdgcn_s_wait_asynccnt(0);
#elif __has_builtin(__builtin_amdgcn_global_load_async_to_lds_b128)
    asm volatile("s_wait_asynccnt 0x0" ::: "memory");
#endif
    __syncthreads();

    // grid-stride: element (g + k*T) always has feature f = g % 32, so each
    // lane of a wave owns one fixed feature -> conflict-free LDS search.
    const int g = blockIdx.x * blockDim.x + threadIdx.x;
    const int T = gridDim.x * blockDim.x;
    const int f = g & (FEAT - 1);

    int i = g;
    for (; i <= n - 4 * T; i += 4 * T) {
        process_four(x, out, sq, i, T, f);
    }
    for (; i < n; i += T) {
        process_one(x, out, sq, i, f);
    }
}

extern "C" void kernel_launch(void* const* d_in, const int* in_sizes, int n_in,
                              void* d_out, int out_size, void* d_ws, size_t ws_size,
                              hipStream_t stream) {
    (void)n_in; (void)d_ws; (void)ws_size;
    const float* x = (const float*)d_in[0];       // [256,4096,32] fp32
    const float* q = (const float*)d_in[1];       // [1000,32] fp32, sorted along Q
    float* out     = (float*)d_out;               // [256,4096,32] fp32
    const int n    = in_sizes[0];                 // 33,554,432

    const int threads = 256;
    const int blocks  = 1024;   // 2 resident blocks/WGP @128KB LDS; 128 elems/thread
    QuantilePreprocessing_kernel<<<blocks, threads, LDS_BYTES, stream>>>(x, q, out, n);
}